// OpeningLoss2D_24189255811164
// MI455X (gfx1250) — compile-verified
//
#include <hip/hip_runtime.h>
#include <stdint.h>

typedef float v2f __attribute__((ext_vector_type(2)));
typedef float v8f __attribute__((ext_vector_type(8)));
typedef unsigned int u32x4 __attribute__((ext_vector_type(4)));
typedef int i32x4 __attribute__((ext_vector_type(4)));
typedef int i32x8 __attribute__((ext_vector_type(8)));

#define MAPW     512
#define TS       64      // output tile is 64x64
#define HALO_W   66      // tile + 1 halo each side
#define NTHREADS 256

// Wave32 block reduction using one V_WMMA_F32_16X16X4_F32:
//   A (16x4) = per-lane value in K=0 (lanes 0-15) / K=2 (lanes 16-31), zeros elsewhere
//   B (4x16) = ones  =>  D[m,n] = v_m + v_{m+16} for every n.
// Summing the 8 D VGPRs gives half-wave sums; one shfl_xor(16) completes the wave.
__device__ __forceinline__ float block_reduce_sum_wmma(float v, float* sdata) {
  v2f a; a[0] = v;    a[1] = 0.0f;
  v2f b; b[0] = 1.0f; b[1] = 1.0f;
  v8f c = {};
  v8f d = __builtin_amdgcn_wmma_f32_16x16x4_f32(false, a, false, b, (short)0, c,
                                                false, false);
  float s = ((d[0] + d[1]) + (d[2] + d[3])) + ((d[4] + d[5]) + (d[6] + d[7]));
  s += __shfl_xor(s, 16, 32);            // lanes 0-15 hold rows 0-7, 16-31 rows 8-15
  const int lane = threadIdx.x & 31;
  const int wid  = threadIdx.x >> 5;
  if (lane == 0) sdata[wid] = s;
  __syncthreads();
  float tot = 0.0f;
  if (threadIdx.x == 0) {
    const int nw = (int)(blockDim.x >> 5);
    for (int i = 0; i < nw; ++i) tot += sdata[i];
  }
  return tot;                             // valid on thread 0
}

__global__ __launch_bounds__(NTHREADS)
void opening_mse_kernel(const float* __restrict__ src, float* __restrict__ partial) {
  __shared__ float tile[HALO_W * HALO_W];
  __shared__ float wsum[NTHREADS / 32];

  const int bid  = blockIdx.x;
  const int map  = bid >> 6;              // 64 tiles per 512x512 map
  const int tidx = bid & 63;
  const int ty   = tidx >> 3;
  const int tx   = tidx & 7;
  const int gy0  = ty * TS - 1;           // tile-local row r -> global row gy0 + r
  const int gx0  = tx * TS - 1;

  const float* mapPtr = src + (size_t)map * (MAPW * MAPW);
  const uint32_t ldsBase = (uint32_t)(uintptr_t)(&tile[0]);

  const bool interior = (ty >= 1) && (ty <= 6) && (tx >= 1) && (tx <= 6);

  if (interior) {
    // ---- Tensor Data Mover: one 2D descriptor loads the whole 66x66 tile ----
    if (threadIdx.x < 32) {               // single wave issues the DMA
      const uint64_t ga = (uint64_t)(uintptr_t)(mapPtr + (size_t)(gy0 * MAPW + gx0));
      u32x4 g0;
      g0[0] = 1u;                                            // count=1, user mode
      g0[1] = ldsBase;                                       // lds_addr (bytes)
      g0[2] = (uint32_t)(ga & 0xFFFFFFFFu);                  // global_addr[31:0]
      g0[3] = (uint32_t)((ga >> 32) & 0x01FFFFFFu)           // global_addr[56:32]
              | 0x80000000u;                                 // type=2 ("image")

      const uint32_t td0 = 1u << 20;     // oversized tensor dims: tile is fully
      const uint32_t td1 = 1u << 20;     // in-bounds, so OOB never triggers
      i32x8 g1 = {};
      g1[0] = (int)(2u << 16);                               // data_size = 4 bytes
      g1[1] = (int)((td0 & 0xFFFFu) << 16);                  // tensor_dim0[15:0]
      g1[2] = (int)(((td0 >> 16) & 0xFFFFu)                  // tensor_dim0[31:16]
                    | ((td1 & 0xFFFFu) << 16));              // tensor_dim1[15:0]
      g1[3] = (int)(((td1 >> 16) & 0xFFFFu)                  // tensor_dim1[31:16]
                    | ((uint32_t)HALO_W << 16));             // tile_dim0 = 66
      g1[4] = (int)HALO_W;                                   // tile_dim1 = 66
      g1[5] = (int)MAPW;                                     // dim0_stride low = 512
      g1[6] = 0;
      g1[7] = 0;
      i32x4 gz = {};
#if __has_include(<hip/amd_detail/amd_gfx1250_TDM.h>)
      i32x8 gz8 = {};
      __builtin_amdgcn_tensor_load_to_lds(g0, g1, gz, gz, gz8, 0);
#else
      __builtin_amdgcn_tensor_load_to_lds(g0, g1, gz, gz, 0);
#endif
      __builtin_amdgcn_s_wait_tensorcnt(0);
    }
  } else {
    // ---- Edge tiles: replicate-clamped halo via async global->LDS loads ----
    for (int e = threadIdx.x; e < HALO_W * HALO_W; e += NTHREADS) {
      int r = e / HALO_W;
      int c = e - r * HALO_W;
      int gy = gy0 + r; gy = gy < 0 ? 0 : (gy > MAPW - 1 ? MAPW - 1 : gy);
      int gx = gx0 + c; gx = gx < 0 ? 0 : (gx > MAPW - 1 ? MAPW - 1 : gx);
      uint32_t goff  = (uint32_t)((gy * MAPW + gx) * 4);
      uint32_t laddr = ldsBase + (uint32_t)(e * 4);
      asm volatile("global_load_async_to_lds_b32 %0, %1, %2"
                   :: "v"(laddr), "v"(goff), "s"(mapPtr) : "memory");
    }
    asm volatile("s_wait_asynccnt 0" ::: "memory");
  }
  __syncthreads();

  // --- Compute: each thread walks one column (C) over 16 output rows ---
  // opened(R,C) needs tile rows R..R+2, cols C..C+2 (tile coords incl. halo).
  const int C  = threadIdx.x & (TS - 1);
  const int R0 = (threadIdx.x >> 6) << 4;       // 0,16,32,48

  float sum  = 0.0f;
  float hm0p = 0.0f, hm1p = 0.0f, mp = 0.0f, x1p = 0.0f;
  int base = R0 * HALO_W + C;

#pragma unroll
  for (int k = 0; k < 18; ++k) {                // rows R0 .. R0+17
    float x0 = tile[base + 0];
    float x1 = tile[base + 1];
    float x2 = tile[base + 2];
    float hm0 = fminf(x0, x1);                  // horizontal 2-min at col C+1
    float hm1 = fminf(x1, x2);                  // horizontal 2-min at col C+2
    if (k >= 1) {
      float er0 = fminf(hm0, hm0p);             // erosion at (r, C+1)
      float er1 = fminf(hm1, hm1p);             // erosion at (r, C+2)
      float m   = fmaxf(er0, er1);              // horizontal dilation step
      if (k >= 2) {
        float opened = fmaxf(mp, m);            // output row R = R0 + k - 2
        float dlt = x1p - opened;               // center = X(R+1, C+1) = prev x1
        sum = fmaf(dlt, dlt, sum);
      }
      mp = m;
    }
    hm0p = hm0; hm1p = hm1; x1p = x1;
    base += HALO_W;
  }

  float bsum = block_reduce_sum_wmma(sum, wsum);
  if (threadIdx.x == 0) partial[bid] = bsum;
}

__global__ __launch_bounds__(NTHREADS)
void final_reduce_kernel(const float* __restrict__ partial, int n,
                         float* __restrict__ out, float invN) {
  __shared__ float wsum[NTHREADS / 32];
  float s = 0.0f;
  for (int i = threadIdx.x; i < n; i += NTHREADS) s += partial[i];
  float tot = block_reduce_sum_wmma(s, wsum);
  if (threadIdx.x == 0) out[0] = tot * invN;
}

extern "C" void kernel_launch(void* const* d_in, const int* in_sizes, int n_in,
                              void* d_out, int out_size, void* d_ws, size_t ws_size,
                              hipStream_t stream) {
  const float* src = (const float*)d_in[0];
  const long long nElem = (long long)in_sizes[0];          // 16*8*512*512
  const int maps   = (int)(nElem / (MAPW * MAPW));         // 128
  const int blocks = maps * 64;                            // 8 x 8 tiles per map

  float* partial = (float*)d_ws;                           // blocks * 4 bytes

  opening_mse_kernel<<<blocks, NTHREADS, 0, stream>>>(src, partial);
  final_reduce_kernel<<<1, NTHREADS, 0, stream>>>(partial, blocks, (float*)d_out,
                                                  1.0f / (float)nElem);
}